// AxonalConnections_62646392980191
// MI455X (gfx1250) — compile-verified
//
#include <hip/hip_runtime.h>

typedef __attribute__((ext_vector_type(2))) float v2f;
typedef __attribute__((ext_vector_type(8))) float v8f;

#define GRID_H 128
#define GRID_W 128
#define NPIX   (GRID_H * GRID_W)   // 16384
#define BATCH  16
#define RAD    3                   // (KERNEL-1)/2 for KERNEL=7, STRIDE=1, PAD=3, DIL=1

// One wave32 computes C[16 batch x 16 targets] for targets (ti, tj0..tj0+15).
// Only the conv band of adjacency (7 source rows x 24 source cols) is read:
// all other entries are exact stored zeros -> traffic 1.07GB -> ~11MB.
//
// K-chunks start at even columns (tj0-4 + 4*ch), so each (sj, sj+1) pair is
// entirely in-range or entirely out-of-range: loads are unconditional aligned
// b64 from a clamped address; only the A values are masked (v_cndmask), which
// zeroes the product without masking B and without any exec-mask branching.
__global__ __launch_bounds__(256)
void AxonalConnections_62646392980191_kernel(const float* __restrict__ spikes,
                                             const float* __restrict__ adj,
                                             float* __restrict__ out)
{
    const int lane        = threadIdx.x & 31;
    const int waveInBlock = threadIdx.x >> 5;
    const int tile        = blockIdx.x * 8 + waveInBlock;  // 1024 tiles
    const int ti          = tile >> 3;                     // target row 0..127
    const int tj0         = (tile & 7) << 4;               // target col base (x16)

    const int idx   = lane & 15;   // M (batch) for A-gather, N (target) for B-gather
    const int half  = lane >> 4;   // 0 -> K{0,1}, 1 -> K{2,3}
    const int kbase = half << 1;

    // Per-chunk pair column, validity and clamped (always-safe) column.
    // Independent of si -> hoisted out of the row loop.
    int  sjp[6];
    bool vld[6];
    #pragma unroll
    for (int ch = 0; ch < 6; ++ch) {
        const int sj0 = tj0 - 4 + (ch << 2) + kbase;       // even
        const bool v  = ((unsigned)sj0 < (unsigned)GRID_W); // covers sj0 and sj0+1
        vld[ch] = v;
        sjp[ch] = v ? sj0 : 0;
    }

    // Adjacency row for this lane's target column n = idx
    const float* adjRow = adj + (size_t)(ti * GRID_W + tj0 + idx) * NPIX;

    v8f c = {0.f, 0.f, 0.f, 0.f, 0.f, 0.f, 0.f, 0.f};

    #pragma unroll
    for (int dsi = -RAD; dsi <= RAD; ++dsi) {
        const int si = ti + dsi;                 // wave-uniform: EXEC stays full
        if (si < 0 || si >= GRID_H) continue;

        const float* spRow  = spikes + idx * NPIX + si * GRID_W;  // batch m = idx
        const float* adjSeg = adjRow + si * GRID_W;

        #pragma unroll
        for (int ch = 0; ch < 6; ++ch) {
            // Unconditional aligned 8B loads from clamped addresses.
            const v2f asrc = *(const v2f*)(spRow  + sjp[ch]);
            const v2f bsrc = *(const v2f*)(adjSeg + sjp[ch]);

            v2f a, b;
            a.x = vld[ch] ? asrc.x : 0.0f;   // A[m][kbase]   (masked)
            a.y = vld[ch] ? asrc.y : 0.0f;   // A[m][kbase+1] (masked)
            b.x = bsrc.x;                    // B[kbase][n]   (A mask kills product)
            b.y = bsrc.y;                    // B[kbase+1][n]

            // D = A(16x4,f32) * B(4x16,f32) + C  -> v_wmma_f32_16x16x4_f32
            c = __builtin_amdgcn_wmma_f32_16x16x4_f32(
                    /*neg_a=*/false, a, /*neg_b=*/false, b,
                    /*c_mod=*/(short)0, c, /*reuse_a=*/false, /*reuse_b=*/false);
        }
    }

    // D layout: VGPR r -> M = r + 8*half, N = idx
    const int t = ti * GRID_W + tj0 + idx;
    float* outCol = out + t;
    #pragma unroll
    for (int r = 0; r < 8; ++r) {
        outCol[(size_t)(r + half * 8) * NPIX] = c[r];
    }
}

extern "C" void kernel_launch(void* const* d_in, const int* in_sizes, int n_in,
                              void* d_out, int out_size, void* d_ws, size_t ws_size,
                              hipStream_t stream) {
    const float* spikes = (const float*)d_in[0];  // [16,128,128] f32
    const float* adj    = (const float*)d_in[1];  // [16384,16384] f32
    float*       out    = (float*)d_out;          // [16,128,128] f32

    // 1024 tiles, 1 wave each; 8 waves (256 threads) per block -> 128 blocks.
    hipLaunchKernelGGL(AxonalConnections_62646392980191_kernel,
                       dim3(128), dim3(256), 0, stream, spikes, adj, out);
}